// ListwiseLoss_3427383902893
// MI455X (gfx1250) — compile-verified
//
#include <hip/hip_runtime.h>
#include <stdint.h>

// ListwiseLoss for MI455X (gfx1250, wave32).
// One wave == one row (H=32, lane==horse). Stable argsort-by-rank done as a
// counting sort using per-rank ballot words in LDS (ranks are in [1,32] so
// bucket b = rank-1 maps 1:1 onto lanes). Input streaming uses the CDNA5
// async global->LDS engine (ASYNCcnt) with per-wave double buffering.
// KL terms are accumulated per-lane across rows; a single butterfly reduction
// runs once per wave at the end (not once per row).
// No WMMA: the workload has no matmul structure; a 16x16x4 f32 WMMA row-sum
// still needs 7 adds + 1 permute to finish a 32-lane sum, no cheaper than the
// 10-op butterfly, so the matrix pipe cannot help here.

#define LW_H            32
#define ROWS_PER_TILE   16
#define WAVES_PER_BLOCK 4
#define THREADS         (WAVES_PER_BLOCK * 32)

// per-wave LDS layout (byte offsets)
#define BUF_SCORES 0
#define BUF_RANKS  2048
#define BUF_MASK   4096
#define BUF_BYTES  4608                  // one tile: 16 rows x (128B + 128B + 32B)
#define WAVE_HIST  (2 * BUF_BYTES)       // 32 x u32 ballot words
#define WAVE_QBUF  (WAVE_HIST + 128)     // 32 x f32 rank-ordered probs
#define WAVE_BYTES (WAVE_QBUF + 128)     // 9472 B per wave
#define SMEM_BYTES (WAVES_PER_BLOCK * WAVE_BYTES)  // 37888 B per block -> 8 blocks/WGP

__device__ __forceinline__ void async_tile_to_lds(
    unsigned ldsBuf, const float* scores, const int* ranks,
    const uint8_t* mask, int tile, int lane)
{
  unsigned long long sB = (unsigned long long)(uintptr_t)scores;
  unsigned long long rB = (unsigned long long)(uintptr_t)ranks;
  unsigned long long mB = (unsigned long long)(uintptr_t)mask;
  unsigned so  = (unsigned)tile * (ROWS_PER_TILE * LW_H * 4u);  // f32/i32 tile byte offset
  unsigned mo  = (unsigned)tile * (ROWS_PER_TILE * LW_H);       // u8 tile byte offset
  unsigned l16 = (unsigned)lane * 16u;
  // 16 rows x 128B = 2048B per stream -> 4 x b128 (512B/wave-instr); mask = 512B -> 1 x b128
#pragma unroll
  for (int i = 0; i < 4; ++i) {
    unsigned go = so + (unsigned)i * 512u + l16;
    unsigned lo = ldsBuf + BUF_SCORES + (unsigned)i * 512u + l16;
    asm volatile("global_load_async_to_lds_b128 %0, %1, %2"
                 :: "v"(lo), "v"(go), "s"(sB) : "memory");
  }
#pragma unroll
  for (int i = 0; i < 4; ++i) {
    unsigned go = so + (unsigned)i * 512u + l16;
    unsigned lo = ldsBuf + BUF_RANKS + (unsigned)i * 512u + l16;
    asm volatile("global_load_async_to_lds_b128 %0, %1, %2"
                 :: "v"(lo), "v"(go), "s"(rB) : "memory");
  }
  {
    unsigned go = mo + l16;
    unsigned lo = ldsBuf + BUF_MASK + l16;
    asm volatile("global_load_async_to_lds_b128 %0, %1, %2"
                 :: "v"(lo), "v"(go), "s"(mB) : "memory");
  }
}

__device__ __forceinline__ void process_row(
    float score, int rank, int maskb, int lane,
    volatile unsigned* hist, volatile float* qbuf,
    float& klLane, float& cntRows)
{
  bool valid = (maskb != 0) && (rank > 0);
  unsigned vm = (unsigned)__ballot(valid);     // wave32: low 32 bits
  int nv = __popc(vm);
  if (nv <= 1) return;                         // row skipped (uniform branch, rare)

  // softmax (T=1); invalid slots are exactly 0, matching -1e9 underflow.
  float e = valid ? __expf(score) : 0.0f;
  float s = e;
#pragma unroll
  for (int d = 16; d > 0; d >>= 1) s += __shfl_xor(s, d, 32);
  float p = e * __builtin_amdgcn_rcpf(s);

  // --- stable counting sort by (rank, lane): ranks in [1,32] -> bucket b=rank-1 in [0,31]
  hist[lane] = 0u;
  int b = valid ? (rank - 1) : 0;
  if (valid) atomicOr((unsigned*)&hist[b], 1u << lane);
  unsigned word = hist[lane];                  // ballot of lanes with rank == lane+1
  unsigned c = (unsigned)__popc(word);
  unsigned x = c;                              // inclusive prefix over buckets (5 steps)
#pragma unroll
  for (int d = 1; d < 32; d <<= 1) {
    unsigned y = __shfl_up(x, (unsigned)d, 32);
    if (lane >= d) x += y;
  }
  unsigned excl = x - c;                       // #valid with rank < lane+1
  unsigned exb = __shfl(excl, b, 32);          // #valid with rank < mine
  unsigned wb  = __shfl(word, b, 32);          // ties mask for my rank value
  unsigned lt  = (1u << lane) - 1u;
  int jrank = (int)(exb + (unsigned)__popc(wb & lt));  // stable sort position
  if (valid) qbuf[jrank] = p;                  // scatter p into rank order
  int jpos = __popc(vm & lt);                  // my position among valid (index order)
  float qv = qbuf[jpos];                       // q paired with my pred prob (in-bounds always)

  // KL term: sum_j q_j(log q_j - log(pr_j + eps)); one per-lane term, no per-row
  // reduction -- lanes accumulate privately, wave reduces once at kernel end.
  float term = valid ? (p * __logf(p) - qv * __logf(p + 1e-8f)) : 0.0f;
  klLane  += term;
  cntRows += 1.0f;                             // uniform in all lanes
}

__global__ __launch_bounds__(THREADS)
void lw_kernel(const float* __restrict__ scores, const int* __restrict__ ranks,
               const uint8_t* __restrict__ mask, float* acc, int rows)
{
  __shared__ unsigned char smem[SMEM_BYTES];
  const int lane = threadIdx.x & 31;
  const int wib  = threadIdx.x >> 5;
  const int wid  = (int)blockIdx.x * WAVES_PER_BLOCK + wib;
  const int nw   = (int)gridDim.x * WAVES_PER_BLOCK;

  // low 32 bits of a shared-aperture address == LDS byte address (ISA 10.2)
  unsigned waveLds = (unsigned)(uintptr_t)(&smem[0]) + (unsigned)wib * WAVE_BYTES;
  volatile unsigned* hist = (volatile unsigned*)(smem + wib * WAVE_BYTES + WAVE_HIST);
  volatile float*    qbuf = (volatile float*)(smem + wib * WAVE_BYTES + WAVE_QBUF);

  float kl = 0.0f, cnt = 0.0f;                 // per-lane accumulators
  const int ntiles = rows / ROWS_PER_TILE;

  int tile = wid, buf = 0;
  if (tile < ntiles)
    async_tile_to_lds(waveLds, scores, ranks, mask, tile, lane);  // prologue into buf 0

  for (; tile < ntiles; tile += nw) {
    int nxt = tile + nw;
    if (nxt < ntiles) {
      async_tile_to_lds(waveLds + (unsigned)(buf ^ 1) * BUF_BYTES,
                        scores, ranks, mask, nxt, lane);
      asm volatile("s_wait_asynccnt 9" ::: "memory");   // current tile's 9 done; next in flight
    } else {
      asm volatile("s_wait_asynccnt 0" ::: "memory");
    }
    const unsigned char* base = smem + wib * WAVE_BYTES + buf * BUF_BYTES;
    const volatile float*   bs = (const volatile float*)(base + BUF_SCORES);
    const volatile int*     br = (const volatile int*)(base + BUF_RANKS);
    const volatile uint8_t* bm = (const volatile uint8_t*)(base + BUF_MASK);
    for (int rr = 0; rr < ROWS_PER_TILE; ++rr) {
      float sc = bs[rr * LW_H + lane];
      int   rk = br[rr * LW_H + lane];
      int   mb = bm[rr * LW_H + lane];
      process_row(sc, rk, mb, lane, hist, qbuf, kl, cnt);
    }
    buf ^= 1;
  }

  // tail rows (rows % 16) via direct coalesced loads
  for (int r = ntiles * ROWS_PER_TILE + wid; r < rows; r += nw) {
    float sc = scores[r * LW_H + lane];
    int   rk = ranks[r * LW_H + lane];
    int   mb = mask[r * LW_H + lane];
    process_row(sc, rk, mb, lane, hist, qbuf, kl, cnt);
  }

  // single wave-level reduction of per-lane KL partials (cnt is lane-uniform)
#pragma unroll
  for (int d = 16; d > 0; d >>= 1) kl += __shfl_xor(kl, d, 32);
  if (lane == 0) {
    atomicAdd(&acc[0], kl);
    atomicAdd(&acc[1], cnt);
  }
}

__global__ void lw_finalize(const float* __restrict__ acc, float* __restrict__ out)
{
  out[0] = acc[0] / fmaxf(acc[1], 1.0f);   // reduction='mean', n_rows = max(count, 1)
}

extern "C" void kernel_launch(void* const* d_in, const int* in_sizes, int n_in,
                              void* d_out, int out_size, void* d_ws, size_t ws_size,
                              hipStream_t stream)
{
  const float*   scores = (const float*)d_in[0];
  const int*     ranks  = (const int*)d_in[1];
  const uint8_t* mask   = (const uint8_t*)d_in[2];   // jnp bool -> 1 byte/elem
  float* out = (float*)d_out;
  float* acc = (float*)d_ws;                          // [0]=kl sum, [1]=row count

  int rows = in_sizes[0] / LW_H;

  hipMemsetAsync(acc, 0, 2 * sizeof(float), stream);  // deterministic accumulators
  int blocks = 2048;                                  // 8192 waves, ~4 tiles each
  lw_kernel<<<blocks, THREADS, 0, stream>>>(scores, ranks, mask, acc, rows);
  lw_finalize<<<1, 1, 0, stream>>>(acc, out);
}